// BNN_SVGD_30855045054885
// MI455X (gfx1250) — compile-verified
//
#include <hip/hip_runtime.h>

// CDNA5 / gfx1250: wave32, V_WMMA_F32_16X16X4_F32 (fp32 matrix core).
typedef __attribute__((ext_vector_type(2))) float v2f;
typedef __attribute__((ext_vector_type(8))) float v8f;
typedef __attribute__((ext_vector_type(4))) int   v4i;
typedef __attribute__((address_space(1))) v4i     gv4i;  // global (AS1)
typedef __attribute__((address_space(3))) v4i     lv4i;  // LDS (AS3)

#define L_W1 2097152   // H*D_IN per particle
#define L_B1 2048
#define L_W2 524288    // D_OUT*H per particle
#define L_B2 256
#define P_TOT 2623744  // total params per particle
#define N_TILE256 10249    // P_TOT / 256 (exact)
#define N_CHUNK32 81992    // P_TOT / 32 (exact)
#define NB_GRAM 512
#define EPS_N 0.003125f    // EPS / n = 0.1 / 32

#if defined(__has_builtin)
#if __has_builtin(__builtin_amdgcn_global_load_async_to_lds_b128)
#define HAVE_ASYNC_LDS 1
#endif
#endif

// Map a 256-wide column tile of the virtual theta matrix to (buffer, per-particle stride).
__device__ __forceinline__ const float* seg256(int tile, const float* W1, const float* b1,
                                               const float* W2, const float* b2, int* stride) {
    if (tile < 8192)  { *stride = L_W1; return W1 + (long)tile * 256; }
    if (tile < 8200)  { *stride = L_B1; return b1 + (long)(tile - 8192) * 256; }
    if (tile < 10248) { *stride = L_W2; return W2 + (long)(tile - 8200) * 256; }
    *stride = L_B2;   return b2;
}

// ---------------------------------------------------------------------------
// Kernel 1: partial Gram. Each block accumulates G_partial = sum over its
// 32x256 tiles of theta_tile * theta_tile^T via fp32 WMMA, LDS-staged.
// Tile loads use async global->LDS (ASYNCcnt) when available, else a staged
// register pipeline, so 8 loads per thread are in flight (not 1).
// ---------------------------------------------------------------------------
__global__ __launch_bounds__(256) void gram_kernel(const float* __restrict__ W1,
                                                   const float* __restrict__ b1,
                                                   const float* __restrict__ W2,
                                                   const float* __restrict__ b2,
                                                   float* __restrict__ Gp) {
    __shared__ alignas(32) float lds[8320];   // 32 rows x pitch 260 (also reused: 8*1024 reduce)
    const int tid  = threadIdx.x;
    const int lane = tid & 31;
    const int w    = tid >> 5;        // wave 0..7 owns k-subrange [w*32, w*32+32)
    const int r    = lane & 15;
    const int g    = lane >> 4;       // lane half: selects K=2g,2g+1 per ISA A/B layout

    v8f c00 = {}, c01 = {}, c10 = {}, c11 = {};

    const int row0 = tid >> 6;           // 0..3
    const int c4   = (tid & 63) << 2;    // 0..252

    for (int tile = blockIdx.x; tile < N_TILE256; tile += gridDim.x) {
        int stride;
        const float* base = seg256(tile, W1, b1, W2, b2, &stride);
#ifdef HAVE_ASYNC_LDS
        // CDNA5 async copy: GLOBAL_LOAD_ASYNC_TO_LDS_B128, tracked by ASYNCcnt.
        #pragma unroll
        for (int p = 0; p < 8; ++p) {
            const int rr = row0 + p * 4;
            __builtin_amdgcn_global_load_async_to_lds_b128(
                (gv4i*)(base + (long)rr * stride + c4),
                (lv4i*)(&lds[rr * 260 + c4]), 0, 0);
        }
#if __has_builtin(__builtin_amdgcn_s_wait_asynccnt)
        __builtin_amdgcn_s_wait_asynccnt(0);
#else
        asm volatile("s_wait_asynccnt 0x0" ::: "memory");
#endif
#else
        // Staged fallback: issue all 8 global loads into distinct registers,
        // then store to LDS -> 8 outstanding loads per thread.
        float4 tmp[8];
        #pragma unroll
        for (int p = 0; p < 8; ++p) {
            const int rr = row0 + p * 4;
            tmp[p] = *(const float4*)(base + (long)rr * stride + c4);
        }
        #pragma unroll
        for (int p = 0; p < 8; ++p) {
            const int rr = row0 + p * 4;
            *(float4*)(&lds[rr * 260 + c4]) = tmp[p];
        }
#endif
        __syncthreads();
        #pragma unroll
        for (int kc = 0; kc < 8; ++kc) {
            const int k0 = w * 32 + kc * 4 + 2 * g;
            // Gram symmetry: per-lane A-frag == B-frag for the same 16-row tile.
            v2f f0 = *(const v2f*)(&lds[r * 260 + k0]);          // particles 0..15
            v2f f1 = *(const v2f*)(&lds[(16 + r) * 260 + k0]);   // particles 16..31
            c00 = __builtin_amdgcn_wmma_f32_16x16x4_f32(false, f0, false, f0, (short)0, c00, false, false);
            c01 = __builtin_amdgcn_wmma_f32_16x16x4_f32(false, f0, false, f1, (short)0, c01, false, false);
            c10 = __builtin_amdgcn_wmma_f32_16x16x4_f32(false, f1, false, f0, (short)0, c10, false, false);
            c11 = __builtin_amdgcn_wmma_f32_16x16x4_f32(false, f1, false, f1, (short)0, c11, false, false);
        }
        __syncthreads();
    }

    // cross-wave reduction in LDS, then one deterministic partial store per block
    *(v8f*)(&lds[w * 1024 + 0 * 256 + lane * 8]) = c00;
    *(v8f*)(&lds[w * 1024 + 1 * 256 + lane * 8]) = c01;
    *(v8f*)(&lds[w * 1024 + 2 * 256 + lane * 8]) = c10;
    *(v8f*)(&lds[w * 1024 + 3 * 256 + lane * 8]) = c11;
    __syncthreads();
    {
        const int sl = tid >> 3;             // source lane
        const int v  = tid & 7;              // source VGPR index
        const int m  = v + 8 * (sl >> 4);    // D layout: row = v + 8*laneGroup
        const int n2 = sl & 15;              // col = lane%16
        #pragma unroll
        for (int t = 0; t < 4; ++t) {
            float s = 0.0f;
            #pragma unroll
            for (int ww = 0; ww < 8; ++ww) s += lds[ww * 1024 + t * 256 + tid];
            const int i = (t >> 1) * 16 + m;
            const int j = (t & 1) * 16 + n2;
            Gp[blockIdx.x * 1024 + i * 32 + j] = s;
        }
    }
}

// ---------------------------------------------------------------------------
// Kernel 2: deterministic reduction of per-block Gram partials.
// ---------------------------------------------------------------------------
__global__ __launch_bounds__(1024) void reduce_kernel(const float* __restrict__ Gp,
                                                      float* __restrict__ G) {
    float s = 0.0f;
    #pragma unroll 8
    for (int b = 0; b < NB_GRAM; ++b) s += Gp[b * 1024 + threadIdx.x];
    G[threadIdx.x] = s;
}

// ---------------------------------------------------------------------------
// Kernel 3: 32x32 coefficient matrix.
// out[i] = sum_j Cf[i][j] * theta[j], with
// Cf[i][j] = (EPS/n)*K[i][j]*[j>=1] + delta_ij*(1 - 3*(EPS/n)*S[i]).
// ---------------------------------------------------------------------------
__global__ __launch_bounds__(1024) void coeff_kernel(const float* __restrict__ G,
                                                     float* __restrict__ Cf) {
    const int i = threadIdx.x >> 5;
    const int j = threadIdx.x & 31;
    const float sqi = G[i * 32 + i];
    const float d2  = sqi + G[j * 32 + j] - 2.0f * G[i * 32 + j];
    const float K   = __expf(-0.5f * fmaxf(d2, 0.0f));
    float S = 0.0f;
    for (int jj = 1; jj < 32; ++jj) {
        const float d = sqi + G[jj * 32 + jj] - 2.0f * G[i * 32 + jj];
        S += __expf(-0.5f * fmaxf(d, 0.0f));
    }
    float c = (j >= 1) ? (EPS_N * K) : 0.0f;
    if (i == j) c += 1.0f - 3.0f * EPS_N * S;
    Cf[i * 32 + j] = c;
}

// ---------------------------------------------------------------------------
// Kernel 4: out = Cf(32x32) @ theta(32xP) via fp32 WMMA, streamed in 32-column
// chunks (full 128B cacheline use). A-frags (coefficients) loaded once per wave.
// ---------------------------------------------------------------------------
__global__ __launch_bounds__(256) void apply_kernel(const float* __restrict__ W1,
                                                    const float* __restrict__ b1,
                                                    const float* __restrict__ W2,
                                                    const float* __restrict__ b2,
                                                    const float* __restrict__ Cf,
                                                    float* __restrict__ out) {
    const int lane = threadIdx.x & 31;
    const int r    = lane & 15;
    const int g    = lane >> 4;
    const int wid  = blockIdx.x * (blockDim.x >> 5) + (threadIdx.x >> 5);
    const int nw   = gridDim.x * (blockDim.x >> 5);

    // A-matrix fragments: A[m][k] = Cf[mt*16+m][k]; lane holds K = kc*4+2g, +1
    v2f a0[8], a1[8];
    #pragma unroll
    for (int kc = 0; kc < 8; ++kc) {
        const int k = kc * 4 + 2 * g;
        a0[kc].x = Cf[r * 32 + k];        a0[kc].y = Cf[r * 32 + k + 1];
        a1[kc].x = Cf[(16 + r) * 32 + k]; a1[kc].y = Cf[(16 + r) * 32 + k + 1];
    }

    for (int c = wid; c < N_CHUNK32; c += nw) {
        const float* base; int stride; long off;
        if (c < 65536)      { base = W1; stride = L_W1; off = (long)c << 5; }
        else if (c < 65600) { base = b1; stride = L_B1; off = (long)(c - 65536) << 5; }
        else if (c < 81984) { base = W2; stride = L_W2; off = (long)(c - 65600) << 5; }
        else                { base = b2; stride = L_B2; off = (long)(c - 81984) << 5; }
        const float* p = base + off + r;

        v8f acc00 = {}, acc01 = {}, acc10 = {}, acc11 = {};
        #pragma unroll
        for (int kc = 0; kc < 8; ++kc) {
            const int krow = kc * 4 + 2 * g;
            const float* pk  = p + (long)krow * stride;
            const float* pk1 = pk + stride;
            // B[k][n] = theta[k][col]: b.x from particle krow, b.y from krow+1
            v2f bl; bl.x = pk[0];  bl.y = pk1[0];    // cols col0 + r
            v2f bh; bh.x = pk[16]; bh.y = pk1[16];   // cols col0 + 16 + r
            acc00 = __builtin_amdgcn_wmma_f32_16x16x4_f32(false, a0[kc], false, bl, (short)0, acc00, false, false);
            acc10 = __builtin_amdgcn_wmma_f32_16x16x4_f32(false, a1[kc], false, bl, (short)0, acc10, false, false);
            acc01 = __builtin_amdgcn_wmma_f32_16x16x4_f32(false, a0[kc], false, bh, (short)0, acc01, false, false);
            acc11 = __builtin_amdgcn_wmma_f32_16x16x4_f32(false, a1[kc], false, bh, (short)0, acc11, false, false);
        }

        const long col0 = (long)c << 5;
        #pragma unroll
        for (int v = 0; v < 8; ++v) {
            const int m = v + 8 * g;  // D layout: row = v + 8*laneGroup, col = lane%16
            out[(long)m        * P_TOT + col0 + r]      = acc00[v];
            out[(long)m        * P_TOT + col0 + 16 + r] = acc01[v];
            out[(long)(16 + m) * P_TOT + col0 + r]      = acc10[v];
            out[(long)(16 + m) * P_TOT + col0 + 16 + r] = acc11[v];
        }
    }
}

extern "C" void kernel_launch(void* const* d_in, const int* in_sizes, int n_in,
                              void* d_out, int out_size, void* d_ws, size_t ws_size,
                              hipStream_t stream) {
    const float* W1 = (const float*)d_in[0];
    const float* b1 = (const float*)d_in[1];
    const float* W2 = (const float*)d_in[2];
    const float* b2 = (const float*)d_in[3];
    // d_in[4] (X) and d_in[5] (y) are unused by the reference.
    float* out = (float*)d_out;

    float* Gp = (float*)d_ws;              // NB_GRAM * 1024 floats (2 MB) partials
    float* G  = Gp + NB_GRAM * 1024;       // 1024 floats: Gram matrix
    float* Cf = G + 1024;                  // 1024 floats: coefficient matrix

    gram_kernel<<<NB_GRAM, 256, 0, stream>>>(W1, b1, W2, b2, Gp);
    reduce_kernel<<<1, 1024, 0, stream>>>(Gp, G);
    coeff_kernel<<<1, 1024, 0, stream>>>(G, Cf);
    apply_kernel<<<1024, 256, 0, stream>>>(W1, b1, W2, b2, Cf, out);
}